// SRLEmbeddings_13331578487024
// MI455X (gfx1250) — compile-verified
//
#include <hip/hip_runtime.h>

typedef __attribute__((ext_vector_type(2))) float v2f;
typedef __attribute__((ext_vector_type(8))) float v8f;

#define NB 32    // batch
#define NS 8     // sentences
#define NT 128   // seq len (T)
#define ND 768   // embedding dim
#define NA 8     // num args
#define NL 8     // arg length
#define NT_PER_WAVE 12   // 48 N-tiles split across 4 wave-pairs

#define OFF_PRED ((unsigned)(NB * NS * ND))                 // after mean output
#define SET_SZ   ((unsigned)(NB * NS * NA * ND))            // size of one id-set output
#define SENTINEL 0xFFFFFFFFu

// One workgroup per (b,s). Builds a 32x128 "count" matrix in LDS
// (rows 0-7 predicate, 8-15 arg0, 16-23 arg1, 24 mean, 25-31 zero),
// then computes Cnt[32x128] @ Emb[128x768] with V_WMMA_F32_16X16X4_F32.
__global__ __launch_bounds__(256) void srl_embed_kernel(
    const float* __restrict__ emb,   // [B,S,T,D]
    const int*   __restrict__ sids,  // [B,S,T]
    const int*   __restrict__ pred,  // [B,S,A,L]
    const int*   __restrict__ a0,    // [B,S,A,L]
    const int*   __restrict__ a1,    // [B,S,A,L]
    float*       __restrict__ out)
{
    __shared__ __align__(16) float cnt[32][NT];   // 16 KB
    __shared__ int sid_s[NT];
    __shared__ unsigned rowtab[32];               // row -> flat offset in out (col 0)

    const int bs  = blockIdx.x;     // 0..B*S-1
    const int tid = threadIdx.x;    // 0..255

    // ---- zero count matrix, stage sentence ids, build row->offset table ----
    for (int i = tid; i < 32 * NT; i += 256) (&cnt[0][0])[i] = 0.0f;
    if (tid < NT) sid_s[tid] = sids[bs * NT + tid];
    if (tid < 32) {
        unsigned off;
        if (tid < 24)        // rows 0-23: pred/arg0/arg1, arg index = tid&7
            off = OFF_PRED + (unsigned)(tid >> 3) * SET_SZ
                + ((unsigned)bs * NA + (unsigned)(tid & 7)) * ND;
        else if (tid == 24)  // mean row
            off = (unsigned)bs * ND;
        else                 // padding rows
            off = SENTINEL;
        rowtab[tid] = off;
    }
    __syncthreads();

    // ---- build scaled-count rows 0..23 (one thread per (set,arg) row) ----
    if (tid < 24) {
        const int set = tid >> 3, a = tid & 7;
        const int* ids = (set == 0) ? pred : (set == 1) ? a0 : a1;
        const int* p = ids + (bs * NA + a) * NL;
        int v[NL];
        int nv = 0;
        #pragma unroll
        for (int l = 0; l < NL; ++l) { v[l] = p[l]; nv += (v[l] != 0); }
        const float inv = 1.0f / (float)(nv > 0 ? nv : 1);
        #pragma unroll 1
        for (int l = 0; l < NL; ++l) {
            if (v[l] != 0) {
                #pragma unroll 1
                for (int t = 0; t < NT; ++t) {          // id -> position
                    if (sid_s[t] == v[l]) { cnt[tid][t] += inv; break; }
                }
            }
        }
    }
    // ---- mean row 24 ----
    if (tid >= 128) cnt[24][tid - 128] = 1.0f / (float)NT;
    __syncthreads();

    // ---- matmul phase: 8 waves, wave w -> M-tile (w&1), 12 N-tiles ----
    const int w    = tid >> 5;
    const int lane = tid & 31;
    const int half = lane >> 4;     // 0: lanes 0-15, 1: lanes 16-31
    const int lm   = lane & 15;
    const int mt   = w & 1;
    const int nt0  = (w >> 1) * NT_PER_WAVE;

    const float* eb = emb + (size_t)bs * NT * ND;
    const int arow = mt * 16 + lm;

    v8f acc[NT_PER_WAVE];
    #pragma unroll
    for (int j = 0; j < NT_PER_WAVE; ++j) {
        v8f z = {0.f, 0.f, 0.f, 0.f, 0.f, 0.f, 0.f, 0.f};
        acc[j] = z;
    }

    for (int k = 0; k < NT; k += 4) {
        const int ka = k + half * 2;                 // this lane's K pair
        // A fragment: VGPR r holds A[M=lane%16][ka+r]  (ISA 32-bit A 16x4 layout)
        const v2f afrag = *(const v2f*)&cnt[arow][ka];   // 8B-aligned LDS load
        #pragma unroll
        for (int j = 0; j < NT_PER_WAVE; ++j) {
            const int nb = (nt0 + j) * 16;
            // B fragment: VGPR r holds B[K=ka+r][N=lane%16]
            v2f bfrag;
            bfrag.x = eb[(size_t)ka * ND + nb + lm];
            bfrag.y = eb[(size_t)(ka + 1) * ND + nb + lm];
            acc[j] = __builtin_amdgcn_wmma_f32_16x16x4_f32(
                false, afrag, false, bfrag, (short)0, acc[j], false, false);
        }
    }

    // ---- store via row->offset table (C/D layout: M = 16*mt + 8*half + v) ----
    unsigned offv[8];
    #pragma unroll
    for (int v = 0; v < 8; ++v) offv[v] = rowtab[mt * 16 + half * 8 + v];

    #pragma unroll
    for (int v = 0; v < 8; ++v) {
        const unsigned off = offv[v];
        const bool valid = (off != SENTINEL);
        #pragma unroll
        for (int j = 0; j < NT_PER_WAVE; ++j) {
            if (valid) out[(size_t)off + (unsigned)((nt0 + j) * 16 + lm)] = acc[j][v];
        }
    }
}

extern "C" void kernel_launch(void* const* d_in, const int* in_sizes, int n_in,
                              void* d_out, int out_size, void* d_ws, size_t ws_size,
                              hipStream_t stream) {
    const float* emb  = (const float*)d_in[0];
    const int*   sids = (const int*)d_in[1];
    const int*   pred = (const int*)d_in[2];
    const int*   a0   = (const int*)d_in[3];
    const int*   a1   = (const int*)d_in[4];
    float* out = (float*)d_out;

    srl_embed_kernel<<<NB * NS, 256, 0, stream>>>(emb, sids, pred, a0, a1, out);
}